// NoisyQuantumModel_54425825575156
// MI455X (gfx1250) — compile-verified
//
#include <hip/hip_runtime.h>
#include <hip/hip_bf16.h>

typedef __attribute__((ext_vector_type(2))) float v2f;
typedef __attribute__((ext_vector_type(8))) float v8f;

#define BATCH   256
#define IN_DIM  784
#define HID     256
#define NQ      6
#define DM_SIZE 4096   // 64x64
#define P1Q_F   0.001f
#define P2Q_F   0.01f
#define QT      128    // threads per quantum block (4 waves)

// ---------------------------------------------------------------------------
// Stage 1: h = relu(x @ W1^T + b1) via fp32 WMMA (V_WMMA_F32_16X16X4_F32)
// grid (16,4), 1 wave/block; each wave computes a 16-row x 64-col strip.
// A layout (16x4 f32): lane&15 = M, VGPR0/1 = K = 2*(lane>>4) + {0,1}
// B layout (4x16 f32): lane&15 = N, VGPR0/1 = K = 2*(lane>>4) + {0,1}
// D layout: VGPR v -> M = v + 8*(lane>>4), N = lane&15
// ---------------------------------------------------------------------------
__global__ __launch_bounds__(32) void gemm1_wmma(const float* __restrict__ x,
                                                 const float* __restrict__ W1,
                                                 const float* __restrict__ bias1,
                                                 float* __restrict__ h) {
  const int lane = threadIdx.x;
  const int m0 = blockIdx.x * 16;
  const int n0 = blockIdx.y * 64;
  const int lr   = lane & 15;
  const int koff = (lane >> 4) * 2;

  const float* xr = x  + (m0 + lr) * IN_DIM + koff;
  const float* w0 = W1 + (n0 + lr) * IN_DIM + koff;
  const float* w1 = w0 + 16 * IN_DIM;
  const float* w2 = w0 + 32 * IN_DIM;
  const float* w3 = w0 + 48 * IN_DIM;

  v8f acc0 = {}, acc1 = {}, acc2 = {}, acc3 = {};
#pragma unroll 4
  for (int k = 0; k < IN_DIM; k += 4) {
    v2f a;   a.x   = xr[k]; a.y   = xr[k + 1];
    v2f b0v; b0v.x = w0[k]; b0v.y = w0[k + 1];
    v2f b1v; b1v.x = w1[k]; b1v.y = w1[k + 1];
    v2f b2v; b2v.x = w2[k]; b2v.y = w2[k + 1];
    v2f b3v; b3v.x = w3[k]; b3v.y = w3[k + 1];
    acc0 = __builtin_amdgcn_wmma_f32_16x16x4_f32(false, a, false, b0v, (short)0, acc0, false, false);
    acc1 = __builtin_amdgcn_wmma_f32_16x16x4_f32(false, a, false, b1v, (short)0, acc1, false, false);
    acc2 = __builtin_amdgcn_wmma_f32_16x16x4_f32(false, a, false, b2v, (short)0, acc2, false, false);
    acc3 = __builtin_amdgcn_wmma_f32_16x16x4_f32(false, a, false, b3v, (short)0, acc3, false, false);
  }
  const int lh = lane >> 4;
#pragma unroll
  for (int v = 0; v < 8; ++v) {
    int row = m0 + v + 8 * lh;
    float* hr = h + row * HID;
    int c = n0 + lr;
    hr[c]      = fmaxf(acc0[v] + bias1[c],      0.f);
    hr[c + 16] = fmaxf(acc1[v] + bias1[c + 16], 0.f);
    hr[c + 32] = fmaxf(acc2[v] + bias1[c + 32], 0.f);
    hr[c + 48] = fmaxf(acc3[v] + bias1[c + 48], 0.f);
  }
}

// ---------------------------------------------------------------------------
// Stage 2: LayerNorm -> (256->6) -> tanh, one block per row
// ---------------------------------------------------------------------------
__global__ __launch_bounds__(256) void encoder_tail(const float* __restrict__ h,
                                                    const float* __restrict__ ln_g,
                                                    const float* __restrict__ ln_b,
                                                    const float* __restrict__ W2,
                                                    const float* __restrict__ b2,
                                                    float* __restrict__ z) {
  __shared__ float red[256];
  const int row = blockIdx.x, t = threadIdx.x;
  float v = h[row * HID + t];
  red[t] = v; __syncthreads();
  for (int s = 128; s > 0; s >>= 1) { if (t < s) red[t] += red[t + s]; __syncthreads(); }
  float mu = red[0] * (1.f / HID);
  __syncthreads();
  float d = v - mu;
  red[t] = d * d; __syncthreads();
  for (int s = 128; s > 0; s >>= 1) { if (t < s) red[t] += red[t + s]; __syncthreads(); }
  float var = red[0] * (1.f / HID);
  __syncthreads();
  float nv = d * rsqrtf(var + 1e-5f) * ln_g[t] + ln_b[t];
  for (int i = 0; i < NQ; ++i) {
    red[t] = nv * W2[i * HID + t]; __syncthreads();
    for (int s = 128; s > 0; s >>= 1) { if (t < s) red[t] += red[t + s]; __syncthreads(); }
    if (t == 0) z[row * NQ + i] = tanhf(red[0] + b2[i]);
    __syncthreads();
  }
}

// ---------------------------------------------------------------------------
// Stage 3: density-matrix simulator, rho resident in LDS (32 KB / block)
// Row/col index: 6 bits, wire 0 = MSB -> qubit q uses bit (5-q).
// ---------------------------------------------------------------------------
__device__ __forceinline__ float2 cmul (float2 a, float2 b) { return make_float2(a.x*b.x - a.y*b.y, a.x*b.y + a.y*b.x); }
__device__ __forceinline__ float2 cmulc(float2 a, float2 b) { return make_float2(a.x*b.x + a.y*b.y, a.y*b.x - a.x*b.y); } // a*conj(b)
__device__ __forceinline__ float2 cadd (float2 a, float2 b) { return make_float2(a.x + b.x, a.y + b.y); }
__device__ __forceinline__ int insert0(int x, int pos) { return ((x >> pos) << (pos + 1)) | (x & ((1 << pos) - 1)); }

// rho -> U rho U^dagger, fused 2x2 block transform (1024 independent blocks)
__device__ void apply_1q(float2* dm, int q, float2 u00, float2 u01, float2 u10, float2 u11) {
  const int pos = 5 - q, m = 1 << pos;
  for (int t = threadIdx.x; t < 1024; t += QT) {
    int r0 = insert0(t >> 5, pos), c0 = insert0(t & 31, pos);
    int i00 = (r0 << 6) | c0;
    int i01 = i00 | m, i10 = i00 | (m << 6), i11 = i00 | m | (m << 6);
    float2 m00 = dm[i00], m01 = dm[i01], m10 = dm[i10], m11 = dm[i11];
    float2 t00 = cadd(cmul(u00, m00), cmul(u01, m10));
    float2 t01 = cadd(cmul(u00, m01), cmul(u01, m11));
    float2 t10 = cadd(cmul(u10, m00), cmul(u11, m10));
    float2 t11 = cadd(cmul(u10, m01), cmul(u11, m11));
    dm[i00] = cadd(cmulc(t00, u00), cmulc(t01, u01));
    dm[i01] = cadd(cmulc(t00, u10), cmulc(t01, u11));
    dm[i10] = cadd(cmulc(t10, u00), cmulc(t11, u01));
    dm[i11] = cadd(cmulc(t10, u10), cmulc(t11, u11));
  }
  __syncthreads();
}

// (1-p)rho + p/3 (XrhoX + YrhoY + ZrhoZ) collapses to an affine 2x2 block map
__device__ void depol(float2* dm, int q, float p) {
  const int pos = 5 - q, m = 1 << pos;
  const float ka = 1.f - 2.f * p / 3.f, kb = 2.f * p / 3.f, kc = 1.f - 4.f * p / 3.f;
  for (int t = threadIdx.x; t < 1024; t += QT) {
    int r0 = insert0(t >> 5, pos), c0 = insert0(t & 31, pos);
    int i00 = (r0 << 6) | c0;
    int i01 = i00 | m, i10 = i00 | (m << 6), i11 = i00 | m | (m << 6);
    float2 m00 = dm[i00], m11 = dm[i11];
    dm[i00] = make_float2(ka * m00.x + kb * m11.x, ka * m00.y + kb * m11.y);
    dm[i11] = make_float2(ka * m11.x + kb * m00.x, ka * m11.y + kb * m00.y);
    float2 m01 = dm[i01], m10 = dm[i10];
    dm[i01] = make_float2(kc * m01.x, kc * m01.y);
    dm[i10] = make_float2(kc * m10.x, kc * m10.y);
  }
  __syncthreads();
}

// CNOT rho CNOT = index permutation (involution) -> pairwise LDS swap
__device__ void cnot_g(float2* dm, int ctrl, int tgt) {
  const int mc = 1 << (5 - ctrl), mt = 1 << (5 - tgt);
  for (int idx = threadIdx.x; idx < DM_SIZE; idx += QT) {
    int r = idx >> 6, c = idx & 63;
    int pr = (r & mc) ? (r ^ mt) : r;
    int pc = (c & mc) ? (c ^ mt) : c;
    int pidx = (pr << 6) | pc;
    if (pidx > idx) { float2 tmp = dm[idx]; dm[idx] = dm[pidx]; dm[pidx] = tmp; }
  }
  __syncthreads();
}

// PennyLane Rot(phi,theta,omega) = RZ(omega) RY(theta) RZ(phi)
__device__ __forceinline__ void rot_u(float phi, float theta, float omega,
                                      float2& u00, float2& u01, float2& u10, float2& u11) {
  float ct = cosf(0.5f * theta), st = sinf(0.5f * theta);
  float sp, cp, sm, cm;
  sincosf(-0.5f * (phi + omega), &sp, &cp);  // ep = e^{-i(phi+omega)/2}
  sincosf( 0.5f * (phi - omega), &sm, &cm);  // em = e^{+i(phi-omega)/2}
  u00 = make_float2( cp * ct,  sp * ct);
  u01 = make_float2(-cm * st, -sm * st);
  u10 = make_float2( cm * st, -sm * st);     // conj(em)*s
  u11 = make_float2( cp * ct, -sp * ct);     // conj(ep)*c
}

__global__ __launch_bounds__(QT) void quantum_kernel(const float* __restrict__ z,
                                                     const float* __restrict__ shared_w,
                                                     const float* __restrict__ task_w,
                                                     const float* __restrict__ Wp,
                                                     const float* __restrict__ bp,
                                                     float* __restrict__ out) {
  __shared__ float2 dm[DM_SIZE];
  __shared__ float probs[64];
  __shared__ float evs[NQ];
  const int b = blockIdx.x;
  const int tid = threadIdx.x;

  for (int i = tid; i < DM_SIZE; i += QT) dm[i] = make_float2(0.f, 0.f);
  __syncthreads();
  if (tid == 0) dm[0] = make_float2(1.f, 0.f);
  __syncthreads();

  // Encoding: RY(z_i * pi) + 1q depolarizing per qubit
  for (int i = 0; i < NQ; ++i) {
    float th = z[b * NQ + i] * 3.14159265358979323846f;
    float c = cosf(0.5f * th), s = sinf(0.5f * th);
    apply_1q(dm, i, make_float2(c, 0.f), make_float2(-s, 0.f),
                    make_float2(s, 0.f), make_float2(c, 0.f));
    depol(dm, i, P1Q_F);
  }
  // Shared StronglyEntanglingLayers (3 layers)
  for (int l = 0; l < 3; ++l) {
    for (int i = 0; i < NQ; ++i) {
      const float* w = shared_w + (l * NQ + i) * 3;
      float2 u00, u01, u10, u11;
      rot_u(w[0], w[1], w[2], u00, u01, u10, u11);
      apply_1q(dm, i, u00, u01, u10, u11);
    }
    int r = (l % (NQ - 1)) + 1;
    for (int i = 0; i < NQ; ++i) cnot_g(dm, i, (i + r) % NQ);
  }
  // Mid-circuit noise
  for (int i = 0; i < NQ; ++i) depol(dm, i, P1Q_F);
  for (int i = 0; i < NQ - 1; ++i) depol(dm, i, P2Q_F);
  // Task StronglyEntanglingLayers (2 layers)
  for (int l = 0; l < 2; ++l) {
    for (int i = 0; i < NQ; ++i) {
      const float* w = task_w + (l * NQ + i) * 3;
      float2 u00, u01, u10, u11;
      rot_u(w[0], w[1], w[2], u00, u01, u10, u11);
      apply_1q(dm, i, u00, u01, u10, u11);
    }
    int r = (l % (NQ - 1)) + 1;
    for (int i = 0; i < NQ; ++i) cnot_g(dm, i, (i + r) % NQ);
  }
  // Final noise
  for (int i = 0; i < NQ; ++i) depol(dm, i, P1Q_F);

  // <Z_i> from diagonal, then 6->2 head
  if (tid < 64) probs[tid] = dm[tid * 64 + tid].x;
  __syncthreads();
  if (tid < NQ) {
    float s = 0.f;
    for (int d = 0; d < 64; ++d)
      s += probs[d] * (((d >> (5 - tid)) & 1) ? -1.f : 1.f);
    evs[tid] = s;
  }
  __syncthreads();
  if (tid < 2) {
    float o = bp[tid];
    for (int i = 0; i < NQ; ++i) o += evs[i] * Wp[tid * NQ + i];
    out[b * 2 + tid] = o;
  }
}

// ---------------------------------------------------------------------------
extern "C" void kernel_launch(void* const* d_in, const int* in_sizes, int n_in,
                              void* d_out, int out_size, void* d_ws, size_t ws_size,
                              hipStream_t stream) {
  const float* x        = (const float*)d_in[0];
  const float* W1       = (const float*)d_in[1];
  const float* b1       = (const float*)d_in[2];
  const float* ln_g     = (const float*)d_in[3];
  const float* ln_b     = (const float*)d_in[4];
  const float* W2       = (const float*)d_in[5];
  const float* b2       = (const float*)d_in[6];
  const float* shared_w = (const float*)d_in[7];
  const float* task_w   = (const float*)d_in[8];
  const float* Wp       = (const float*)d_in[9];
  const float* bp       = (const float*)d_in[10];
  float* out = (float*)d_out;

  float* h = (float*)d_ws;            // 256*256 floats
  float* z = h + BATCH * HID;         // 256*6 floats

  gemm1_wmma  <<<dim3(16, 4), 32,  0, stream>>>(x, W1, b1, h);
  encoder_tail<<<BATCH,       256, 0, stream>>>(h, ln_g, ln_b, W2, b2, z);
  quantum_kernel<<<BATCH,     QT,  0, stream>>>(z, shared_w, task_w, Wp, bp, out);
}